// Autocorrelation_47674136986073
// MI455X (gfx1250) — compile-verified
//
#include <hip/hip_runtime.h>
#include <math.h>

// ---------------------------------------------------------------------------
// Autocorrelation attention (FFT-free): proj GEMM (WMMA f32) -> per-(b,d)
// circular correlation + top-13 + softmax + rolled-V aggregation -> broadcast
// to 8 identical heads.
// Shapes fixed by the reference: B=16, L=1024, D=512, DH=64, ktop=13.
// ---------------------------------------------------------------------------

typedef __attribute__((ext_vector_type(2))) float v2f;
typedef __attribute__((ext_vector_type(8))) float v8f;

#define NB   16
#define NL   1024
#define ND   512
#define NDH  64
#define KTOP 13

// ---------------------------------------------------------------------------
// Kernel 1: projections  xT[b][d][l] = (X[b,l,:] . Wq[:,d]) + bq[d]
// One wave per 16x16 output tile, K swept 4 at a time with
// V_WMMA_F32_16X16X4_F32 (fp32 kept end-to-end: top-k is rank sensitive).
// A frag (16x4 f32): lane m=lane&15; VGPR v holds K = 2*(lane>>4)+v.
// B frag (4x16 f32): lane n=lane&15; VGPR v holds K = 2*(lane>>4)+v.
// C/D (16x16 f32):   VGPR r holds M = r + 8*(lane>>4), N = lane&15.
// ---------------------------------------------------------------------------
__global__ void __launch_bounds__(128)
ac_proj_wmma(const float* __restrict__ Q, const float* __restrict__ K,
             const float* __restrict__ V, const float* __restrict__ Wq,
             const float* __restrict__ bq,
             float* __restrict__ qT, float* __restrict__ kT,
             float* __restrict__ vT)
{
    const float* X; float* Y;
    if      (blockIdx.z == 0) { X = Q; Y = qT; }
    else if (blockIdx.z == 1) { X = K; Y = kT; }
    else                      { X = V; Y = vT; }

    const int wid   = threadIdx.x >> 5;      // wave -> N tile (DH/16 = 4)
    const int lane  = threadIdx.x & 31;
    const int m0    = blockIdx.x * 16;       // tile row in flattened [B*L]
    const int n0    = wid * 16;
    const int mrow  = m0 + (lane & 15);
    const int khalf = (lane >> 4) * 2;       // 0 or 2
    const int ncol  = n0 + (lane & 15);

    const float* __restrict__ arow = X + (size_t)mrow * ND;

    v8f c = {};
    for (int k0 = 0; k0 < ND; k0 += 4) {
        v2f a, b;
        const float2 av = *(const float2*)(arow + k0 + khalf);   // 8B aligned
        a[0] = av.x;
        a[1] = av.y;
        b[0] = Wq[(k0 + khalf + 0) * NDH + ncol];
        b[1] = Wq[(k0 + khalf + 1) * NDH + ncol];
        c = __builtin_amdgcn_wmma_f32_16x16x4_f32(
                /*neg_a=*/false, a, /*neg_b=*/false, b,
                /*c_mod=*/(short)0, c, /*reuse_a=*/false, /*reuse_b=*/false);
    }

    const float bias  = bq[ncol];
    const int   b_idx = m0 >> 10;            // m0 / L
    const int   l0    = m0 & (NL - 1);
    const int   mhi   = (lane >> 4) * 8;
    float* __restrict__ yrow =
        Y + ((size_t)(b_idx * NDH + ncol)) * NL + l0 + mhi;
#pragma unroll
    for (int r = 0; r < 8; ++r)
        yrow[r] = c[r] + bias;
}

// ---------------------------------------------------------------------------
// Kernel 2: per (b,d) signal — fused circular correlation, top-13 (ties ->
// lowest index, matching lax.top_k), softmax over lags, rolled-V aggregate.
// All in LDS (4 x 4KB signals + reduction scratch ~ 18.6 KB << 320 KB WGP).
// aggT aliases kT: each block loads its own kT slice into LDS before the
// aliased write, and blocks touch disjoint slices.
// ---------------------------------------------------------------------------
__global__ void __launch_bounds__(256)
ac_corr_topk_agg(const float* __restrict__ qT, const float* __restrict__ kT,
                 const float* __restrict__ vT, float* __restrict__ aggT)
{
    __shared__ float qs[NL];
    __shared__ float ks[NL];
    __shared__ float vs[NL];
    __shared__ float cr[NL];
    __shared__ float rv[256];
    __shared__ int   ri[256];
    __shared__ float topv[KTOP];
    __shared__ int   topi[KTOP];
    __shared__ float w[KTOP];

    const int b   = blockIdx.x;
    const int d   = blockIdx.y;
    const int tid = threadIdx.x;
    const size_t base = ((size_t)(b * NDH + d)) * NL;

#pragma unroll
    for (int j = 0; j < 4; ++j) {
        const int i = tid + 256 * j;
        qs[i] = qT[base + i];
        ks[i] = kT[base + i];
        vs[i] = vT[base + i];
    }
    __syncthreads();

    // corr[tau] = sum_l q[(l+tau)%L] * k[l]
    // ks[l] is a broadcast read; qs[(l+tau)&1023] is stride-1 across lanes.
#pragma unroll
    for (int j = 0; j < 4; ++j) {
        const int tau = tid + 256 * j;
        float s = 0.0f;
        for (int l = 0; l < NL; ++l)
            s = fmaf(qs[(l + tau) & (NL - 1)], ks[l], s);
        cr[tau] = s;
    }
    __syncthreads();

    // top-13 by repeated argmax with masking; tie-break = lowest index.
    for (int t = 0; t < KTOP; ++t) {
        float bv = -3.402823466e38f;
        int   bi = NL;
#pragma unroll
        for (int j = 0; j < 4; ++j) {
            const int i = tid + 256 * j;      // ascending -> lowest-idx tie win
            const float v = cr[i];
            if (v > bv) { bv = v; bi = i; }
        }
        rv[tid] = bv; ri[tid] = bi;
        __syncthreads();
        for (int s = 128; s > 0; s >>= 1) {
            if (tid < s) {
                const float ov = rv[tid + s]; const int oi = ri[tid + s];
                if (ov > rv[tid] || (ov == rv[tid] && oi < ri[tid])) {
                    rv[tid] = ov; ri[tid] = oi;
                }
            }
            __syncthreads();
        }
        if (tid == 0) {
            topv[t] = rv[0];
            topi[t] = ri[0];
            cr[ri[0]] = -3.402823466e38f;     // mask selected lag
        }
        __syncthreads();
    }

    // softmax over the 13 lag scores (topv[0] is the max by construction)
    if (tid == 0) {
        const float m = topv[0];
        float sum = 0.0f;
        for (int t = 0; t < KTOP; ++t) { w[t] = __expf(topv[t] - m); sum += w[t]; }
        const float inv = 1.0f / sum;
        for (int t = 0; t < KTOP; ++t) w[t] *= inv;
    }
    __syncthreads();

    // agg[l] = sum_k w[k] * v[(l + idx_k) % L]   (roll by -idx)
#pragma unroll
    for (int j = 0; j < 4; ++j) {
        const int l = tid + 256 * j;
        float acc = 0.0f;
#pragma unroll
        for (int t = 0; t < KTOP; ++t)
            acc = fmaf(w[t], vs[(l + topi[t]) & (NL - 1)], acc);
        aggT[base + l] = acc;
    }
}

// ---------------------------------------------------------------------------
// Kernel 3: LDS transpose + 8x head broadcast with coalesced output rows:
// out[b, l, h*64 + d] = aggT[b, d, l]   for h = 0..7
// ---------------------------------------------------------------------------
__global__ void __launch_bounds__(256)
ac_writeout(const float* __restrict__ aggT, float* __restrict__ out)
{
    __shared__ float tile[NDH][33];          // 33 -> conflict-free transpose
    const int b   = blockIdx.x;
    const int l0  = blockIdx.y * 32;
    const int tid = threadIdx.x;

#pragma unroll
    for (int i = 0; i < 8; ++i) {            // 64 d x 32 l = 2048 loads
        const int flat = i * 256 + tid;
        const int d  = flat >> 5;
        const int ll = flat & 31;
        tile[d][ll] = aggT[((size_t)(b * NDH + d)) * NL + l0 + ll];
    }
    __syncthreads();

#pragma unroll
    for (int i = 0; i < 64; ++i) {           // 32 l x 512 c = 16384 stores
        const int flat = i * 256 + tid;
        const int ll = flat >> 9;
        const int c  = flat & 511;           // c = h*64 + d -> tile[c&63]
        out[((size_t)(b * NL + l0 + ll)) * ND + c] = tile[c & 63][ll];
    }
}

// ---------------------------------------------------------------------------
extern "C" void kernel_launch(void* const* d_in, const int* in_sizes, int n_in,
                              void* d_out, int out_size, void* d_ws, size_t ws_size,
                              hipStream_t stream) {
    const float* Q  = (const float*)d_in[0];
    const float* K  = (const float*)d_in[1];
    const float* V  = (const float*)d_in[2];
    const float* Wq = (const float*)d_in[3];
    const float* bq = (const float*)d_in[4];
    float* out = (float*)d_out;

    // Workspace: qT | kT | vT, each B*DH*L floats (4 MB) = 12 MB total.
    const size_t plane = (size_t)NB * NDH * NL;
    float* qT = (float*)d_ws;
    float* kT = qT + plane;
    float* vT = kT + plane;
    float* aggT = kT;   // safe alias: kernel 2 LDS-stages kT before overwrite

    // 1) projections: (B*L/16) M-tiles x 4 N-tiles (one wave each) x {Q,K,V}
    ac_proj_wmma<<<dim3((NB * NL) / 16, 1, 3), 128, 0, stream>>>(
        Q, K, V, Wq, bq, qT, kT, vT);

    // 2) fused correlation / top-13 / softmax / aggregate, one block per (b,d)
    ac_corr_topk_agg<<<dim3(NB, NDH), 256, 0, stream>>>(qT, kT, vT, aggT);

    // 3) transpose + head broadcast into [B, L, 512]
    ac_writeout<<<dim3(NB, NL / 32), 256, 0, stream>>>(aggT, out);
}